// HetConv_31920196944464
// MI455X (gfx1250) — compile-verified
//
#include <hip/hip_runtime.h>
#include <stdint.h>

#define DIM 256
#define OUT_DIM 512
#define WAVES_PER_BLOCK 8
#define CHUNK 32

// ---------- CDNA5 async-to-LDS + wait helpers (probe-verified on this toolchain) ----------
#if __has_builtin(__builtin_amdgcn_s_wait_asynccnt)
#define WAIT_ASYNC(n) __builtin_amdgcn_s_wait_asynccnt(n)
#else
#define WAIT_ASYNC(n) asm volatile("s_wait_asynccnt %0" ::"i"(n) : "memory")
#endif

__device__ __forceinline__ void async_load_b32(const void* gptr, void* ldsptr) {
#if __has_builtin(__builtin_amdgcn_global_load_async_to_lds_b32)
  __builtin_amdgcn_global_load_async_to_lds_b32(
      (__attribute__((address_space(1))) int*)gptr,
      (__attribute__((address_space(3))) int*)ldsptr, /*offset=*/0, /*cpol=*/0);
#else
  uint32_t loff = (uint32_t)(uintptr_t)ldsptr;
  asm volatile("global_load_async_to_lds_b32 %0, %1, off" ::"v"(loff), "v"(gptr)
               : "memory");
#endif
}

// Hardware f32 atomic add (no-return). Prefer the explicit builtin so we are
// guaranteed global_atomic_add_f32 rather than a CAS loop.
__device__ __forceinline__ void atomic_add_f32(float* p, float v) {
#if __has_builtin(__builtin_amdgcn_global_atomic_fadd_f32)
  (void)__builtin_amdgcn_global_atomic_fadd_f32(
      (__attribute__((address_space(1))) float*)p, v);
#else
  __hip_atomic_fetch_add(p, v, __ATOMIC_RELAXED, __HIP_MEMORY_SCOPE_AGENT);
#endif
}

// ---------- COO SpMM scatter: both adjacencies in one edge index space ----------
// Per wave: stream 32-edge chunks (row/col/val) into LDS via the async engine
// (double-buffered on ASYNCcnt), then per edge gather the 1KB x-row with
// 2x b128 loads/lane and fire 8 no-return global_atomic_add_f32/lane.
__global__ __launch_bounds__(256) void spmm_scatter_kernel(
    const float* __restrict__ x,
    const int* __restrict__ row1, const int* __restrict__ col1,
    const float* __restrict__ val1,
    const int* __restrict__ row2, const int* __restrict__ col2,
    const float* __restrict__ val2,
    float* __restrict__ out, int E) {
  __shared__ int lds[2][WAVES_PER_BLOCK][3][CHUNK];

  const int lane   = threadIdx.x & 31;
  const int wave   = threadIdx.x >> 5;
  const int gwave  = blockIdx.x * WAVES_PER_BLOCK + wave;
  const int nwave  = gridDim.x * WAVES_PER_BLOCK;
  const int totalE = 2 * E;
  const int stride = nwave * CHUNK;

  auto issue = [&](int chunkBase, int buf) {
    int ee = chunkBase + lane;
    if (ee < totalE) {
      const int *rp, *cp;
      const float* vp;
      int idx;
      if (ee < E) { rp = row1; cp = col1; vp = val1; idx = ee; }
      else        { rp = row2; cp = col2; vp = val2; idx = ee - E; }
      async_load_b32(rp + idx, &lds[buf][wave][0][lane]);
      async_load_b32(cp + idx, &lds[buf][wave][1][lane]);
      async_load_b32(vp + idx, &lds[buf][wave][2][lane]);
    }
  };

  int base = gwave * CHUNK;
  if (base >= totalE) return;

  int buf = 0;
  issue(base, 0);  // prologue: 3 async ops outstanding

  for (int chunk = base; chunk < totalE; chunk += stride) {
    int nxt = chunk + stride;
    if (nxt < totalE) {
      issue(nxt, buf ^ 1);  // 6 outstanding; wait until only the new 3 remain
      WAIT_ASYNC(3);
    } else {
      WAIT_ASYNC(0);
    }
    asm volatile("" ::: "memory");  // keep LDS reads below the wait

    int count = totalE - chunk;
    if (count > CHUNK) count = CHUNK;

    for (int j = 0; j < count; ++j) {
      int rr   = lds[buf][wave][0][j];
      int cc   = lds[buf][wave][1][j];
      float vv = __int_as_float(lds[buf][wave][2][j]);

      // prefetch the gather row two edges ahead (global_prefetch_b8)
      if (j + 2 < count) {
        int cpf = lds[buf][wave][1][j + 2];
        __builtin_prefetch(x + cpf * DIM + lane * 8, 0, 0);
      }

      const float4* xr = (const float4*)(x + cc * DIM);
      float4 a = xr[lane];
      float4 b = xr[lane + 32];

      int oo = ((chunk + j) < E) ? 0 : DIM;  // adjacency 1 -> cols [0,256), 2 -> [256,512)
      float* orow = out + rr * OUT_DIM + oo;

      atomic_add_f32(orow + 4 * lane + 0, vv * a.x);
      atomic_add_f32(orow + 4 * lane + 1, vv * a.y);
      atomic_add_f32(orow + 4 * lane + 2, vv * a.z);
      atomic_add_f32(orow + 4 * lane + 3, vv * a.w);
      atomic_add_f32(orow + 128 + 4 * lane + 0, vv * b.x);
      atomic_add_f32(orow + 128 + 4 * lane + 1, vv * b.y);
      atomic_add_f32(orow + 128 + 4 * lane + 2, vv * b.z);
      atomic_add_f32(orow + 128 + 4 * lane + 3, vv * b.w);
    }
    buf ^= 1;
  }
}

// ---------- zero-init output (harness poisons d_out; atomics accumulate into it) ----------
__global__ void zero_out_kernel(float4* __restrict__ out, int n4) {
  int i = blockIdx.x * blockDim.x + threadIdx.x;
  if (i < n4) out[i] = make_float4(0.f, 0.f, 0.f, 0.f);
}

extern "C" void kernel_launch(void* const* d_in, const int* in_sizes, int n_in,
                              void* d_out, int out_size, void* d_ws, size_t ws_size,
                              hipStream_t stream) {
  const float* x    = (const float*)d_in[0];
  const int*   row1 = (const int*)d_in[1];
  const int*   col1 = (const int*)d_in[2];
  const float* val1 = (const float*)d_in[3];
  const int*   row2 = (const int*)d_in[4];
  const int*   col2 = (const int*)d_in[5];
  const float* val2 = (const float*)d_in[6];
  float* out = (float*)d_out;

  const int E = in_sizes[1];  // edges per adjacency

  int n4 = out_size / 4;
  zero_out_kernel<<<(n4 + 255) / 256, 256, 0, stream>>>((float4*)out, n4);

  // Exactly 2 chunks (64 edges) per wave: prologue stages chunk A, the loop
  // stages chunk B while processing A, then drains B. E=320000 -> 1250 blocks.
  int chunks = (2 * E + CHUNK - 1) / CHUNK;
  int waves  = (chunks + 1) / 2;
  int blocks = (waves + WAVES_PER_BLOCK - 1) / WAVES_PER_BLOCK;
  if (blocks < 1) blocks = 1;
  spmm_scatter_kernel<<<blocks, 256, 0, stream>>>(x, row1, col1, val1, row2,
                                                  col2, val2, out, E);
}